// Net_23484881175023
// MI455X (gfx1250) — compile-verified
//
#include <hip/hip_runtime.h>
#include <hip/hip_bf16.h>

typedef float v2f __attribute__((ext_vector_type(2)));
typedef float v4f __attribute__((ext_vector_type(4)));
typedef float v8f __attribute__((ext_vector_type(8)));

#define BATCH 512
#define LIN   24576
#define LP    8192   // LIN/3
#define HID   10

// ---------- unified nonlinearity: out = s * base(m*x) + d ----------
#if __has_builtin(__builtin_amdgcn_tanhf)
#define NL_M_SIG  0.5f
#define NL_S_SIG  0.5f
#define NL_D_SIG  0.5f
#define NL_M_TANH 1.0f
#define NL_S_TANH 1.0f
#define NL_D_TANH 0.0f
__device__ __forceinline__ float nl(float x, float m, float s, float d) {
  return fmaf(s, __builtin_amdgcn_tanhf(m * x), d);   // v_tanh_f32 (TRANS)
}
#else
#define NL_M_SIG  (-1.442695041f)
#define NL_S_SIG  1.0f
#define NL_D_SIG  0.0f
#define NL_M_TANH (-2.885390082f)
#define NL_S_TANH 2.0f
#define NL_D_TANH (-1.0f)
__device__ __forceinline__ float nl(float x, float m, float s, float d) {
  return fmaf(s, __builtin_amdgcn_rcpf(1.0f + __builtin_amdgcn_exp2f(m * x)), d);
}
#endif

// ---------- half-wave broadcast pair: lo_b[lane]=g[lane&15], hi_b[lane]=g[(lane&15)+16] ----------
// v_permlane16_swap_b32 (pure VALU) replaces ds_bpermute in the serial chain when available.
__device__ __forceinline__ void half_pair(float g, int hi, float& lo_b, float& hi_b) {
#if __has_builtin(__builtin_amdgcn_permlane16_swap)
  (void)hi;
  auto p = __builtin_amdgcn_permlane16_swap(__float_as_uint(g), __float_as_uint(g),
                                            false, false);
  lo_b = __uint_as_float(p[0]);   // {gL, gL}
  hi_b = __uint_as_float(p[1]);   // {gH, gH}
#else
  float s = __shfl_xor(g, 16);
  lo_b = hi ? s : g;
  hi_b = hi ? g : s;
#endif
}

// -------- conv1d k=3 stride=3 + relu; output batch-major (BATCH, LP) --------
__global__ void conv3_kernel(const float* __restrict__ x, const float* __restrict__ cw,
                             const float* __restrict__ cb, float* __restrict__ c) {
  const int b = blockIdx.x;                    // 0..511
  const float w0 = cw[0], w1 = cw[1], w2 = cw[2], bb = cb[0];
  const float* xb = x + (size_t)b * LIN;
  float* cr = c + (size_t)b * LP;
  for (int t = threadIdx.x; t < LP; t += blockDim.x) {
    float a0 = xb[3 * t + 0];
    float a1 = xb[3 * t + 1];
    float a2 = xb[3 * t + 2];
    float v = fmaf(a0, w0, fmaf(a1, w1, fmaf(a2, w2, bb)));
    cr[t] = fmaxf(v, 0.0f);
  }
}

// ---------------- LSTM recurrence via V_WMMA_F32_16X16X4_F32 ----------------
// One wave (32 lanes) per 16-batch tile. A = [w_hh | w_ih | bias] (gates x K12),
// B = [h ; conv ; 1]^T (K12 x batch16), D tile: gate = 16j + r + 8*hi, batch = lane&15.
template <bool FUSED>
__global__ __launch_bounds__(32) void lstm_wmma_kernel(
    const float* __restrict__ cbuf,   // (BATCH, LP) batch-major (unused when FUSED)
    const float* __restrict__ x,      // (BATCH, LIN) (used when FUSED)
    const float* __restrict__ convw, const float* __restrict__ convb,
    const float* __restrict__ w_ih,   // (40)
    const float* __restrict__ w_hh,   // (40,10)
    const float* __restrict__ b_ih,   // (40)
    const float* __restrict__ b_hh,   // (40)
    const float* __restrict__ mlp_w,  // (3,10)
    const float* __restrict__ mlp_b,  // (3)
    float* __restrict__ out)          // (BATCH, 3)
{
  const int lane = threadIdx.x;   // 0..31
  const int m16  = lane & 15;
  const int hi   = lane >> 4;     // 0: low half, 1: high half
  const int b0   = blockIdx.x * 16;

  // ---- preload A (constant across all 8192 steps): 3 gate-tiles x 3 K-chunks ----
  v2f A[3][3];
#pragma unroll
  for (int j = 0; j < 3; ++j) {
    const int gate = 16 * j + m16;
    const bool gvb = gate < 40;
    const int gidx = gvb ? gate : 0;
#pragma unroll
    for (int q = 0; q < 3; ++q) {
      const int k0 = 4 * q + 2 * hi;               // K covered by this lane-half
      const int ka = (k0 < HID) ? k0 : 0;
      const int kb = (k0 + 1 < HID) ? k0 + 1 : 0;
      float ax = w_hh[gidx * HID + ka];  ax = (gvb && k0     < HID) ? ax : 0.0f;
      float ay = w_hh[gidx * HID + kb];  ay = (gvb && k0 + 1 < HID) ? ay : 0.0f;
      if (q == 2) {                                 // K=10 -> w_ih, K=11 -> b_ih+b_hh
        float wi = w_ih[gidx];                wi = gvb ? wi : 0.0f;
        float bs = b_ih[gidx] + b_hh[gidx];   bs = gvb ? bs : 0.0f;
        ax = hi ? wi : ax;
        ay = hi ? bs : ay;
      }
      A[j][q] = (v2f){ax, ay};
    }
  }

  // nonlinearity constants for the mixed tile j==1 (gates 16..31; tanh on 20..29)
  float m1[8], s1[8], d1[8];
#pragma unroll
  for (int r = 0; r < 8; ++r) {
    const bool th = hi ? (r <= 5) : (r >= 4);
    m1[r] = th ? NL_M_TANH : NL_M_SIG;
    s1[r] = th ? NL_S_TANH : NL_S_SIG;
    d1[r] = th ? NL_D_TANH : NL_D_SIG;
  }

  // h state lives directly in the B-operand registers (low lanes: k 0,1,4,5,8,9;
  // high lanes: k 2,3,6,7 plus (conv,1.0) at K=10,11)
  v2f hB[3] = {(v2f){0.f, 0.f}, (v2f){0.f, 0.f}, (v2f){0.f, hi ? 1.0f : 0.f}};
  float cc[6] = {0.f, 0.f, 0.f, 0.f, 0.f, 0.f};

  // -------- one LSTM timestep (pure register/VALU work in the chain) --------
  auto lstm_step = [&](float cv) {
    hB[2].x = hi ? cv : hB[2].x;                   // K=10 row carries conv input

    v8f c0 = {}, c1 = {}, c2 = {};
#pragma unroll
    for (int q = 0; q < 3; ++q) {
      c0 = __builtin_amdgcn_wmma_f32_16x16x4_f32(false, A[0][q], false, hB[q], (short)0, c0, false, false);
      c1 = __builtin_amdgcn_wmma_f32_16x16x4_f32(false, A[1][q], false, hB[q], (short)0, c1, false, false);
      c2 = __builtin_amdgcn_wmma_f32_16x16x4_f32(false, A[2][q], false, hB[q], (short)0, c2, false, false);
    }

    float g0[8], g1[8], g2[8];
#pragma unroll
    for (int r = 0; r < 8; ++r) {
      g0[r] = nl(c0[r], NL_M_SIG, NL_S_SIG, NL_D_SIG);  // gates 0..15: sigmoid
      g1[r] = nl(c1[r], m1[r], s1[r], d1[r]);           // gates 16..31: mixed
      g2[r] = nl(c2[r], NL_M_SIG, NL_S_SIG, NL_D_SIG);  // gates 32..39: sigmoid
    }

    // half-wave broadcast pairs (VALU permlane swap; no LDS in the chain)
    float P00l, P00h, P01l, P01h, P02l, P02h, P03l, P03h, P06l, P06h, P07l, P07h;
    float P10l, P10h, P11l, P11h, P14l, P14h, P15l, P15h, P16l, P16h, P17l, P17h;
    float P20l, P20h, P21l, P21h, P24l, P24h, P25l, P25h;
    half_pair(g0[0], hi, P00l, P00h);  half_pair(g0[1], hi, P01l, P01h);
    half_pair(g0[2], hi, P02l, P02h);  half_pair(g0[3], hi, P03l, P03h);
    half_pair(g0[6], hi, P06l, P06h);  half_pair(g0[7], hi, P07l, P07h);
    half_pair(g1[0], hi, P10l, P10h);  half_pair(g1[1], hi, P11l, P11h);
    half_pair(g1[4], hi, P14l, P14h);  half_pair(g1[5], hi, P15l, P15h);
    half_pair(g1[6], hi, P16l, P16h);  half_pair(g1[7], hi, P17l, P17h);
    half_pair(g2[0], hi, P20l, P20h);  half_pair(g2[1], hi, P21l, P21h);
    half_pair(g2[4], hi, P24l, P24h);  half_pair(g2[5], hi, P25l, P25h);

    // gather (i,f,g,o) per hidden slot; slot k: low {0,1,4,5,8,9}, high {2,3,6,7}
    float iv[6], fv[6], gv[6], ov[6];
    iv[0] = hi ? P02l : g0[0];  fv[0] = hi ? g0[4] : P02h;  gv[0] = hi ? P16l : g1[4];  ov[0] = hi ? P20l : P16h;
    iv[1] = hi ? P03l : g0[1];  fv[1] = hi ? g0[5] : P03h;  gv[1] = hi ? P17l : g1[5];  ov[1] = hi ? P21l : P17h;
    iv[2] = hi ? P06l : g0[4];  fv[2] = hi ? P10l : P06h;   gv[2] = hi ? g1[2] : P10h;  ov[2] = hi ? P24l : g2[2];
    iv[3] = hi ? P07l : g0[5];  fv[3] = hi ? P11l : P07h;   gv[3] = hi ? g1[3] : P11h;  ov[3] = hi ? P25l : g2[3];
    iv[4] = P00h;               fv[4] = g1[2];              gv[4] = P14h;               ov[4] = g2[6];  // low-only
    iv[5] = P01h;               fv[5] = g1[3];              gv[5] = P15h;               ov[5] = g2[7];  // low-only

    float hn[6];
#pragma unroll
    for (int s = 0; s < 6; ++s) {
      cc[s] = fmaf(fv[s], cc[s], iv[s] * gv[s]);
      hn[s] = ov[s] * nl(cc[s], NL_M_TANH, NL_S_TANH, NL_D_TANH);
    }
    hB[0] = (v2f){hn[0], hn[1]};
    hB[1] = (v2f){hn[2], hn[3]};
    hB[2].x = hn[4];                    // high lanes overwritten with cv next step
    hB[2].y = hi ? 1.0f : hn[5];        // keep the bias row = 1.0 on high lanes
  };

  if constexpr (!FUSED) {
    // 4 steps of conv input per b128 load, double-buffered 8 steps ahead.
    const float* crow = cbuf + (size_t)(b0 + m16) * LP;
    v4f cva = *(const v4f*)(crow + 0);
    v4f cvb = *(const v4f*)(crow + 4);
    for (int t4 = 0; t4 < LP; t4 += 4) {
      v4f cur = cva;
      cva = cvb;
      if (t4 + 8 < LP) {
        cvb = *(const v4f*)(crow + t4 + 8);
        __builtin_prefetch(crow + t4 + 64, 0, 0);   // global_prefetch ahead of the stream
      }
      lstm_step(cur[0]);
      lstm_step(cur[1]);
      lstm_step(cur[2]);
      lstm_step(cur[3]);
    }
  } else {
    const float cw0 = convw[0], cw1 = convw[1], cw2 = convw[2], cb0 = convb[0];
    const float* xrow = x + (size_t)(b0 + m16) * LIN;
    for (int t4 = 0; t4 < LP; t4 += 4) {
      v4f q0 = *(const v4f*)(xrow + 3 * t4 + 0);
      v4f q1 = *(const v4f*)(xrow + 3 * t4 + 4);
      v4f q2 = *(const v4f*)(xrow + 3 * t4 + 8);
      float c0v = fmaxf(fmaf(q0[0], cw0, fmaf(q0[1], cw1, fmaf(q0[2], cw2, cb0))), 0.0f);
      float c1v = fmaxf(fmaf(q0[3], cw0, fmaf(q1[0], cw1, fmaf(q1[1], cw2, cb0))), 0.0f);
      float c2v = fmaxf(fmaf(q1[2], cw0, fmaf(q1[3], cw1, fmaf(q2[0], cw2, cb0))), 0.0f);
      float c3v = fmaxf(fmaf(q2[1], cw0, fmaf(q2[2], cw1, fmaf(q2[3], cw2, cb0))), 0.0f);
      lstm_step(c0v);
      lstm_step(c1v);
      lstm_step(c2v);
      lstm_step(c3v);
    }
  }

  // ---------------- head: out[b,:] = h_n @ mlp_w.T + mlp_b ----------------
  const int kbase[6] = {0, 1, 4, 5, 8, 9};
  float hv[6] = {hB[0].x, hB[0].y, hB[1].x, hB[1].y, hB[2].x, hB[2].y};
#pragma unroll
  for (int o = 0; o < 3; ++o) {
    float acc = 0.0f;
#pragma unroll
    for (int s = 0; s < 6; ++s) {
      const int k  = kbase[s] + 2 * hi;            // high lanes own k+2; k>=10 invalid
      const int kc = (k < HID) ? k : 0;
      float w = mlp_w[o * HID + kc];
      w = (k < HID) ? w : 0.0f;
      acc = fmaf(hv[s], w, acc);
    }
    acc += __shfl_xor(acc, 16);                    // combine the lane pair (off chain)
    if (lane < 16) out[(b0 + lane) * 3 + o] = acc + mlp_b[o];
  }
}

extern "C" void kernel_launch(void* const* d_in, const int* in_sizes, int n_in,
                              void* d_out, int out_size, void* d_ws, size_t ws_size,
                              hipStream_t stream) {
  const float* x     = (const float*)d_in[0];
  const float* convw = (const float*)d_in[1];
  const float* convb = (const float*)d_in[2];
  const float* w_ih  = (const float*)d_in[3];
  const float* w_hh  = (const float*)d_in[4];
  const float* b_ih  = (const float*)d_in[5];
  const float* b_hh  = (const float*)d_in[6];
  const float* mlp_w = (const float*)d_in[7];
  const float* mlp_b = (const float*)d_in[8];
  float* out  = (float*)d_out;
  float* cbuf = (float*)d_ws;

  const size_t need = (size_t)LP * BATCH * sizeof(float);   // 16 MB scratch
  if (ws_size >= need) {
    conv3_kernel<<<BATCH, 256, 0, stream>>>(x, convw, convb, cbuf);
    lstm_wmma_kernel<false><<<BATCH / 16, 32, 0, stream>>>(
        cbuf, x, convw, convb, w_ih, w_hh, b_ih, b_hh, mlp_w, mlp_b, out);
  } else {
    lstm_wmma_kernel<true><<<BATCH / 16, 32, 0, stream>>>(
        cbuf, x, convw, convb, w_ih, w_hh, b_ih, b_hh, mlp_w, mlp_b, out);
  }
}